// GCNconv_50517405336095
// MI455X (gfx1250) — compile-verified
//
#include <hip/hip_runtime.h>

#define N_NODES 50000
#define N_EDGES 800000
#define IN_DIM  256
#define OUT_DIM 64

typedef __attribute__((ext_vector_type(2))) float v2f;
typedef __attribute__((ext_vector_type(8))) float v8f;

// ---------------------------------------------------------------------------
// init: zero output accumulator, deg = 1.0 (self loop)
// ---------------------------------------------------------------------------
__global__ void k_init(float* __restrict__ out, float* __restrict__ deg) {
    int idx = blockIdx.x * blockDim.x + threadIdx.x;
    if (idx < N_NODES * OUT_DIM) out[idx] = 0.0f;
    if (idx < N_NODES) deg[idx] = 1.0f;
}

// ---------------------------------------------------------------------------
// degree count: deg[dst[e]] += 1  (L2-resident f32 atomics)
// ---------------------------------------------------------------------------
__global__ void k_degree(const int* __restrict__ dst, float* __restrict__ deg) {
    int e = blockIdx.x * blockDim.x + threadIdx.x;
    if (e < N_EDGES) atomicAdd(&deg[dst[e]], 1.0f);
}

// ---------------------------------------------------------------------------
// deg -> d^{-1/2} in place
// ---------------------------------------------------------------------------
__global__ void k_rsqrt(float* __restrict__ deg) {
    int i = blockIdx.x * blockDim.x + threadIdx.x;
    if (i < N_NODES) deg[i] = rsqrtf(deg[i]);
}

// ---------------------------------------------------------------------------
// Pack W [64x256] into WMMA B-fragment order so GEMM inner-loop loads are
// coalesced. For n-tile nt, k-step s, lane l: pair = W[nt*16+(l&15)][4s+2*(l>>4)+{0,1}]
// flat index = ((nt*64 + s)*32 + l)*2
// ---------------------------------------------------------------------------
__global__ void k_packW(const float* __restrict__ W, float* __restrict__ Wf) {
    int idx = blockIdx.x * blockDim.x + threadIdx.x;   // 0..8191
    if (idx >= (OUT_DIM / 16) * (IN_DIM / 4) * 32) return;
    int lane = idx & 31;
    int s    = (idx >> 5) & 63;
    int nt   = idx >> 11;
    int n = nt * 16 + (lane & 15);
    int k = s * 4 + 2 * (lane >> 4);
    Wf[idx * 2 + 0] = W[n * IN_DIM + k];
    Wf[idx * 2 + 1] = W[n * IN_DIM + k + 1];
}

// ---------------------------------------------------------------------------
// H = X @ W^T + b via V_WMMA_F32_16X16X4_F32.
// Block = 128 threads = 4 waves; block computes a 16x64 tile of H.
// Wave w computes the 16x16 tile at columns [16w, 16w+16).
// K = 256 -> 64 WMMA steps. 50000 % 16 == 0 -> no guards, EXEC all ones.
// ---------------------------------------------------------------------------
__global__ __launch_bounds__(128) void k_gemm(const float* __restrict__ X,
                                              const float* __restrict__ Wf,
                                              const float* __restrict__ b,
                                              float* __restrict__ H) {
    // 260-float pitch: LDS bank = (4r + c) % 64 -> conflict-free float2 reads
    __shared__ float tx[16][260];
    const int tile = blockIdx.x;
    const int tid  = threadIdx.x;
    const int w    = tid >> 5;
    const int lane = tid & 31;
    const int r    = lane & 15;
    const int half = lane >> 4;

    // Stage 16x256 X tile into LDS, fully coalesced float4 loads
    const float4* X4 = reinterpret_cast<const float4*>(X + (size_t)tile * 16 * IN_DIM);
    for (int i = tid; i < 16 * (IN_DIM / 4); i += 128) {
        float4 v = X4[i];
        int rr = i >> 6, cc = i & 63;
        *reinterpret_cast<float4*>(&tx[rr][cc * 4]) = v;
    }
    __syncthreads();

    v8f acc = {};
    const float* wf = Wf + (size_t)w * (64 * 32 * 2);
#pragma unroll 8
    for (int s = 0; s < 64; ++s) {
        const int k = s * 4 + 2 * half;                 // A frag: row r, K pair
        float2 av = *reinterpret_cast<const float2*>(&tx[r][k]);
        float2 bv = *reinterpret_cast<const float2*>(wf + ((s * 32 + lane) << 1));
        v2f a;  a.x  = av.x; a.y  = av.y;
        v2f bb; bb.x = bv.x; bb.y = bv.y;
        // D = A(16x4) x B(4x16) + C, f32 in/out
        acc = __builtin_amdgcn_wmma_f32_16x16x4_f32(false, a, false, bb,
                                                    (short)0, acc, false, false);
    }

    // C/D layout: VGPR i -> M = i + 8*half, N = lane&15 (within tile)
    const float bias = b[w * 16 + r];
    const int   mb   = tile * 16 + half * 8;
    const int   col  = w * 16 + r;
#pragma unroll
    for (int i = 0; i < 8; ++i) {
        H[(size_t)(mb + i) * OUT_DIM + col] = acc[i] + bias;
    }
}

// ---------------------------------------------------------------------------
// Edge scatter: out[dst] += H[src] * dinv[src]*dinv[dst]
// One thread per (edge, channel): gathers coalesced, atomics L2-resident.
// ---------------------------------------------------------------------------
__global__ void k_scatter(const float* __restrict__ H, const float* __restrict__ dinv,
                          const int* __restrict__ src, const int* __restrict__ dst,
                          float* __restrict__ out) {
    unsigned idx = blockIdx.x * blockDim.x + threadIdx.x;
    if (idx >= (unsigned)N_EDGES * OUT_DIM) return;
    int e = (int)(idx >> 6);
    int j = (int)(idx & 63);
    int s = src[e], d = dst[e];
    float ne = dinv[s] * dinv[d];
    atomicAdd(&out[(size_t)d * OUT_DIM + j], H[(size_t)s * OUT_DIM + j] * ne);
}

// ---------------------------------------------------------------------------
// Finalize: add self-loop term, ReLU, deterministic hash-based dropout p=0.5
// ---------------------------------------------------------------------------
__device__ __forceinline__ unsigned hash_u32(unsigned x) {
    x ^= 42u * 0x9E3779B1u;
    x ^= x >> 16; x *= 0x85EBCA77u;
    x ^= x >> 13; x *= 0xC2B2AE3Du;
    x ^= x >> 16;
    return x;
}

__global__ void k_finalize(const float* __restrict__ H, const float* __restrict__ dinv,
                           float* __restrict__ out) {
    int idx = blockIdx.x * blockDim.x + threadIdx.x;
    if (idx >= N_NODES * OUT_DIM) return;
    int i = idx >> 6;
    float di = dinv[i];
    float v  = out[idx] + H[idx] * di * di;
    v = fmaxf(v, 0.0f);
    unsigned h = hash_u32((unsigned)idx);
    out[idx] = (h & 0x80000000u) ? 0.0f : v * 2.0f;   // keep w.p. 0.5, scale 1/0.5
}

// ---------------------------------------------------------------------------
extern "C" void kernel_launch(void* const* d_in, const int* in_sizes, int n_in,
                              void* d_out, int out_size, void* d_ws, size_t ws_size,
                              hipStream_t stream) {
    const float* X   = (const float*)d_in[0];
    const float* W   = (const float*)d_in[1];
    const float* b   = (const float*)d_in[2];
    const int*   src = (const int*)d_in[3];
    const int*   dst = (const int*)d_in[4];
    float* out = (float*)d_out;

    // workspace layout (floats): H [3.2M] | deg/dinv [50k] | packed W [16384]
    float* H   = (float*)d_ws;
    float* deg = H + (size_t)N_NODES * OUT_DIM;
    float* Wf  = deg + N_NODES;

    k_init   <<<(N_NODES * OUT_DIM + 255) / 256, 256, 0, stream>>>(out, deg);
    k_degree <<<(N_EDGES + 255) / 256,           256, 0, stream>>>(dst, deg);
    k_rsqrt  <<<(N_NODES + 255) / 256,           256, 0, stream>>>(deg);
    k_packW  <<<(8192 + 255) / 256,              256, 0, stream>>>(W, Wf);
    k_gemm   <<<N_NODES / 16,                    128, 0, stream>>>(X, Wf, b, H);
    {
        unsigned total = (unsigned)N_EDGES * OUT_DIM;          // 51.2M
        k_scatter<<<(total + 255) / 256, 256, 0, stream>>>(H, deg, src, dst, out);
    }
    k_finalize<<<(N_NODES * OUT_DIM + 255) / 256, 256, 0, stream>>>(H, deg, out);
}